// SymGATLayer_26053271617750
// MI455X (gfx1250) — compile-verified
//
#include <hip/hip_runtime.h>

typedef __attribute__((ext_vector_type(2))) float v2f;
typedef __attribute__((ext_vector_type(8))) float v8f;
typedef __attribute__((ext_vector_type(4))) int   v4i;

#define NN 8192      // nodes
#define DD 128       // feature dim
#define NEG_SLOPE 0.2f

// ---- order-preserving float<->uint encoding (monotone) ----
__device__ __forceinline__ unsigned enc_f32(float f) {
  unsigned u = __float_as_uint(f);
  return (u & 0x80000000u) ? ~u : (u | 0x80000000u);
}
__device__ __forceinline__ float dec_f32(unsigned u) {
  return __uint_as_float((u & 0x80000000u) ? (u ^ 0x80000000u) : ~u);
}

// ---------------------------------------------------------------------------
// 0) zero the dense attention buffer (as uint sentinel 0) + init max slot
// ---------------------------------------------------------------------------
__global__ void k_zero(unsigned* __restrict__ S, unsigned* __restrict__ maxslot, long n4) {
  long i = (long)blockIdx.x * blockDim.x + threadIdx.x;
  const long step = (long)gridDim.x * blockDim.x;
  uint4 z = make_uint4(0u, 0u, 0u, 0u);
  for (; i < n4; i += step) ((uint4*)S)[i] = z;
  if (blockIdx.x == 0 && threadIdx.x == 0) *maxslot = 0u;
}

// ---------------------------------------------------------------------------
// 1) h_trans = h @ W^T : wave-per-16x16-tile f32 WMMA GEMM, K=128.
//    B(k,n) = W[n*DD + k] -> both A and B fragments are contiguous b64 loads.
// ---------------------------------------------------------------------------
__global__ void k_linear_gemm(const float* __restrict__ A,   // h [NN x DD]
                              const float* __restrict__ W,   // [DD x DD] (out,in)
                              float* __restrict__ C) {       // h_trans [NN x DD]
  const int lane = threadIdx.x & 31;
  const int wave = threadIdx.x >> 5;
  const int tile = blockIdx.x * (blockDim.x >> 5) + wave;
  const int m0 = (tile / (DD / 16)) * 16;
  const int n0 = (tile % (DD / 16)) * 16;
  const int half = lane >> 4;   // 0: lanes 0-15 (K+0/1), 1: lanes 16-31 (K+2/3)
  const int l16  = lane & 15;

  const float* arow = A + (long)(m0 + l16) * DD + 2 * half;
  const float* brow = W + (long)(n0 + l16) * DD + 2 * half;   // B[k][n]=W[n][k]
  v8f acc = {};
#pragma unroll 8
  for (int k0 = 0; k0 < DD; k0 += 4) {
    v2f a = *(const v2f*)(arow + k0);
    v2f b = *(const v2f*)(brow + k0);
    acc = __builtin_amdgcn_wmma_f32_16x16x4_f32(false, a, false, b,
                                                (short)0, acc, false, false);
  }
#pragma unroll
  for (int r = 0; r < 8; ++r)
    C[(long)(m0 + r + 8 * half) * DD + (n0 + l16)] = acc[r];
}

// ---------------------------------------------------------------------------
// 2) per-node scores: s1[n] = att[0:128].h_trans[n], s2[n] = att[128:256].h_trans[n]
// ---------------------------------------------------------------------------
__global__ void k_node_scores(const float* __restrict__ ht, const float* __restrict__ att,
                              float* __restrict__ s1, float* __restrict__ s2) {
  const int lane = threadIdx.x & 31;
  const int n = blockIdx.x * (blockDim.x >> 5) + (threadIdx.x >> 5);
  const float* row = ht + (long)n * DD;
  float a = 0.f, b = 0.f;
#pragma unroll
  for (int i = 0; i < DD / 32; ++i) {
    int k = lane + 32 * i;
    float v = row[k];
    a = fmaf(v, att[k], a);
    b = fmaf(v, att[DD + k], b);
  }
#pragma unroll
  for (int off = 16; off > 0; off >>= 1) {
    a += __shfl_down(a, off, 32);
    b += __shfl_down(b, off, 32);
  }
  if (lane == 0) { s1[n] = a; s2[n] = b; }
}

// ---------------------------------------------------------------------------
// 3) edge scatter: S[r][c] = max-dup leaky_relu(s1[r]+s2[c]) (ordered-uint),
//    plus global max of edge_e (wave-reduced atomic)
// ---------------------------------------------------------------------------
__global__ void k_scatter(const long long* __restrict__ ei,
                          const float* __restrict__ s1, const float* __restrict__ s2,
                          unsigned* __restrict__ S, unsigned* __restrict__ maxslot, int E) {
  const int e = blockIdx.x * blockDim.x + threadIdx.x;
  unsigned u = 0u;
  if (e < E) {
    int r = (int)ei[e];
    int c = (int)ei[E + e];
    float v = s1[r] + s2[c];
    v = (v >= 0.f) ? v : NEG_SLOPE * v;
    u = enc_f32(v);
    atomicMax(&S[(long)r * NN + c], u);
  }
#pragma unroll
  for (int off = 16; off > 0; off >>= 1)
    u = max(u, (unsigned)__shfl_xor((int)u, off, 32));
  if ((threadIdx.x & 31) == 0) atomicMax(maxslot, u);
}

// ---------------------------------------------------------------------------
// 4) symmetrize (decode ordered-uint -> float), set diagonal to fill value
// ---------------------------------------------------------------------------
__global__ void k_sym_diag(unsigned* __restrict__ Su, float* __restrict__ Sf,
                           const unsigned* __restrict__ maxslot) {
  if (blockIdx.y > blockIdx.x) return;            // upper triangle blocks only
  const int j = blockIdx.x * 16 + threadIdx.x;
  const int i = blockIdx.y * 16 + threadIdx.y;
  if (i > j) return;
  if (i == j) { Sf[(long)i * NN + i] = dec_f32(*maxslot); return; }
  unsigned ua = Su[(long)i * NN + j];
  unsigned ub = Su[(long)j * NN + i];
  float a = ua ? dec_f32(ua) : 0.f;
  float b = ub ? dec_f32(ub) : 0.f;
  float v = 0.5f * (a + b);
  Sf[(long)i * NN + j] = v;
  Sf[(long)j * NN + i] = v;
}

// ---------------------------------------------------------------------------
// 5) row softmax in place; whole 32KB row staged in LDS (single HBM rd+wr)
// ---------------------------------------------------------------------------
__global__ void k_row_softmax(float* __restrict__ S) {
  __shared__ float buf[NN];
  __shared__ float red[256];
  const int tid = threadIdx.x;
  const long base = (long)blockIdx.x * NN;

  float m = -3.402823466e38f;
  for (int j = tid; j < NN; j += 256) { float v = S[base + j]; buf[j] = v; m = fmaxf(m, v); }
  red[tid] = m; __syncthreads();
#pragma unroll
  for (int s = 128; s > 0; s >>= 1) { if (tid < s) red[tid] = fmaxf(red[tid], red[tid + s]); __syncthreads(); }
  m = red[0]; __syncthreads();

  float sum = 0.f;
  for (int j = tid; j < NN; j += 256) { float e = __expf(buf[j] - m); buf[j] = e; sum += e; }
  red[tid] = sum; __syncthreads();
#pragma unroll
  for (int s = 128; s > 0; s >>= 1) { if (tid < s) red[tid] += red[tid + s]; __syncthreads(); }
  const float inv = 1.0f / red[0];

  for (int j = tid; j < NN; j += 256) S[base + j] = buf[j] * inv;
}

// ---------------------------------------------------------------------------
// 6) h_prime = attention(8192x8192) @ h_trans(8192x128), fused ELU.
//    Wave owns a 16x128 output stripe (8 accumulators) -> attention read once.
//    h_trans staged through LDS in 64-row chunks via GLOBAL_LOAD_ASYNC_TO_LDS
//    (ASYNCcnt path, no VGPR round-trip); padded stride 136 so the K and K+2
//    rows used by the two lane-halves land on disjoint bank groups.
// ---------------------------------------------------------------------------
#define KCHUNK 64
#define HS_STRIDE 136

#if defined(__has_builtin)
#if __has_builtin(__builtin_amdgcn_global_load_async_to_lds_b128)
#define HAVE_ASYNC_LDS 1
#endif
#endif

typedef __attribute__((address_space(1))) v4i g_v4i;   // global int4
typedef __attribute__((address_space(3))) v4i l_v4i;   // LDS int4

__global__ void k_attn_gemm_elu(const float* __restrict__ P,   // softmaxed attention
                                const float* __restrict__ H,   // h_trans
                                float* __restrict__ O) {       // h_prime
  __shared__ float Hs[KCHUNK * HS_STRIDE];
  const int lane = threadIdx.x & 31;
  const int wave = threadIdx.x >> 5;
  const int half = lane >> 4;
  const int l16  = lane & 15;
  const int m0 = (blockIdx.x * (blockDim.x >> 5) + wave) * 16;

  v8f acc[8] = {};
  const float* arow = P + (long)(m0 + l16) * NN + 2 * half;

  for (int kc = 0; kc < NN; kc += KCHUNK) {
    __syncthreads();                                   // protect previous chunk reads
#if defined(HAVE_ASYNC_LDS)
    // async copy: wave w copies rows w, w+8, ...; one B128 per lane per row
#pragma unroll
    for (int i = 0; i < KCHUNK / 8; ++i) {
      const int row = wave + i * 8;
      const float* src = H + (long)(kc + row) * DD + lane * 4;
      float* dst = &Hs[row * HS_STRIDE + lane * 4];
      __builtin_amdgcn_global_load_async_to_lds_b128((g_v4i*)src, (l_v4i*)dst, 0, 0);
    }
#if __has_builtin(__builtin_amdgcn_s_wait_asynccnt)
    __builtin_amdgcn_s_wait_asynccnt(0);
#else
    asm volatile("s_wait_asynccnt 0" ::: "memory");
#endif
#else
    for (int idx = threadIdx.x; idx < KCHUNK * (DD / 4); idx += blockDim.x) {
      int row = idx >> 5;            // / (128/4)
      int cq  = idx & 31;
      *(float4*)&Hs[row * HS_STRIDE + cq * 4] =
          *(const float4*)&H[(long)(kc + row) * DD + cq * 4];
    }
#endif
    __syncthreads();

    if (kc + KCHUNK < NN)                              // prefetch next A chunk
      __builtin_prefetch(arow + kc + KCHUNK, 0, 1);

    for (int k0 = 0; k0 < KCHUNK; k0 += 4) {
      v2f a = *(const v2f*)(arow + kc + k0);
      const float* hs0 = &Hs[(k0 + 2 * half) * HS_STRIDE + l16];
#pragma unroll
      for (int t = 0; t < 8; ++t) {
        v2f b;
        b.x = hs0[t * 16];
        b.y = hs0[HS_STRIDE + t * 16];
        acc[t] = __builtin_amdgcn_wmma_f32_16x16x4_f32(false, a, false, b,
                                                       (short)0, acc[t], false, false);
      }
    }
  }

#pragma unroll
  for (int t = 0; t < 8; ++t)
#pragma unroll
    for (int r = 0; r < 8; ++r) {
      float v = acc[t][r];
      v = (v > 0.f) ? v : (__expf(v) - 1.0f);          // ELU
      O[(long)(m0 + r + 8 * half) * DD + (t * 16 + l16)] = v;
    }
}

// ---------------------------------------------------------------------------
extern "C" void kernel_launch(void* const* d_in, const int* in_sizes, int n_in,
                              void* d_out, int out_size, void* d_ws, size_t ws_size,
                              hipStream_t stream) {
  const float*      h   = (const float*)d_in[0];        // [8192,128]
  const long long*  ei  = (const long long*)d_in[1];    // [2,E] int64
  const float*      W   = (const float*)d_in[2];        // [128,128] (out,in)
  const float*      att = (const float*)d_in[3];        // [256]
  const int E = in_sizes[1] / 2;

  // outputs: [h_prime (8192*128) | attention (8192*8192)]
  float*    hprime = (float*)d_out;
  float*    attn   = hprime + (long)NN * DD;
  unsigned* attnu  = (unsigned*)attn;

  // workspace: h_trans | s1 | s2 | maxslot
  float*    ht      = (float*)d_ws;
  float*    s1      = ht + (long)NN * DD;
  float*    s2      = s1 + NN;
  unsigned* maxslot = (unsigned*)(s2 + NN);

  // 0) zero dense attention (uint sentinel) + max slot
  k_zero<<<2048, 256, 0, stream>>>(attnu, maxslot, (long)NN * NN / 4);

  // 1) h_trans = h @ W^T
  {
    int tiles = (NN / 16) * (DD / 16);                 // 4096
    k_linear_gemm<<<tiles / 8, 256, 0, stream>>>(h, W, ht);
  }

  // 2) per-node attention scores
  k_node_scores<<<NN / 8, 256, 0, stream>>>(ht, att, s1, s2);

  // 3) edge scatter + global max
  k_scatter<<<(E + 255) / 256, 256, 0, stream>>>(ei, s1, s2, attnu, maxslot, E);

  // 4) symmetrize + diagonal fill (in place, decode to float)
  {
    dim3 g(NN / 16, NN / 16), b(16, 16);
    k_sym_diag<<<g, b, 0, stream>>>(attnu, attn, maxslot);
  }

  // 5) dense row softmax in place
  k_row_softmax<<<NN, 256, 0, stream>>>(attn);

  // 6) h_prime = softmax(attention) @ h_trans, fused ELU
  k_attn_gemm_elu<<<NN / 128, 256, 0, stream>>>(attn, ht, hprime);
}